// AggregationNetwork_70308614635811
// MI455X (gfx1250) — compile-verified
//
#include <hip/hip_runtime.h>

typedef __attribute__((ext_vector_type(2))) float v2f;
typedef __attribute__((ext_vector_type(8))) float v8f;

#define N_NODES 100000
#define N_EDGES 1280000
#define F_IN 64
#define HID 64

// ---------------- utility ----------------
__global__ void fill_kernel(float* __restrict__ p, float v, int n) {
    int i = blockIdx.x * blockDim.x + threadIdx.x;
    if (i < n) p[i] = v;
}

// deg[col[e]] += w[e]   (deg pre-filled with 1.0 for the self-loop weight)
__global__ void deg_kernel(const int* __restrict__ col, const float* __restrict__ w,
                           float* __restrict__ deg, int e) {
    int i = blockIdx.x * blockDim.x + threadIdx.x;
    if (i < e) atomicAdd(&deg[col[i]], w[i]);
}

// deg -> dinv = rsqrt(deg) in place
__global__ void dinv_kernel(float* __restrict__ deg, int n) {
    int i = blockIdx.x * blockDim.x + threadIdx.x;
    if (i < n) {
        float d = deg[i];
        deg[i] = (d > 0.0f) ? rsqrtf(d) : 0.0f;
    }
}

// norm[e] = dinv[row]*w*dinv[col]   (shared by both conv layers)
__global__ void norm_kernel(const int* __restrict__ row, const int* __restrict__ col,
                            const float* __restrict__ w, const float* __restrict__ dinv,
                            float* __restrict__ norm, int e) {
    int i = blockIdx.x * blockDim.x + threadIdx.x;
    if (i < e) norm[i] = dinv[row[i]] * w[i] * dinv[col[i]];
}

// ---------------- WMMA GEMM: h0[N,64] = x[N,64] @ W1[64,64], fp32 ----------------
// One wave computes a 16-row x 64-col output strip via V_WMMA_F32_16X16X4_F32.
// A 16x4 f32 layout (ISA 7.12.2): lanes 0-15 M=lane hold K=+0,+1 in v0,v1;
//                                 lanes 16-31 hold K=+2,+3.
// B 4x16 / C,D 16x16: rows striped across lanes within a VGPR (half-wave split).
__global__ void gemm_wmma_kernel(const float* __restrict__ x,
                                 const float* __restrict__ W1,
                                 float* __restrict__ h0, int nrows) {
    const int lane  = threadIdx.x & 31;
    const int wavei = threadIdx.x >> 5;
    const int strip = blockIdx.x * (blockDim.x >> 5) + wavei;
    if (strip * 16 >= nrows) return;            // wave-uniform: EXEC stays all-ones

    const int row0 = strip * 16;
    const int m    = lane & 15;                 // M (for A) / N (for B,C,D)
    const int kb   = (lane >> 4) << 1;          // 0 for lanes 0-15, 2 for lanes 16-31

    v8f acc0 = {}, acc1 = {}, acc2 = {}, acc3 = {};
    const float* __restrict__ xrow = x + (size_t)(row0 + m) * F_IN;

#pragma unroll
    for (int k0 = 0; k0 < F_IN; k0 += 4) {
        v2f a;
        a.x = xrow[k0 + kb];
        a.y = xrow[k0 + kb + 1];

        const float* __restrict__ bA = W1 + (size_t)(k0 + kb) * HID + m;       // K = k0+kb
        const float* __restrict__ bB = bA + HID;                               // K = k0+kb+1

        v2f b;
        b.x = bA[0];  b.y = bB[0];
        acc0 = __builtin_amdgcn_wmma_f32_16x16x4_f32(false, a, false, b, (short)0, acc0, false, false);
        b.x = bA[16]; b.y = bB[16];
        acc1 = __builtin_amdgcn_wmma_f32_16x16x4_f32(false, a, false, b, (short)0, acc1, false, false);
        b.x = bA[32]; b.y = bB[32];
        acc2 = __builtin_amdgcn_wmma_f32_16x16x4_f32(false, a, false, b, (short)0, acc2, false, false);
        b.x = bA[48]; b.y = bB[48];
        acc3 = __builtin_amdgcn_wmma_f32_16x16x4_f32(false, a, false, b, (short)0, acc3, false, false);
    }

    // D layout: VGPR j -> M = j (lanes 0-15) / j+8 (lanes 16-31), N = lane&15
    const int half = lane >> 4;
#pragma unroll
    for (int j = 0; j < 8; ++j) {
        size_t r = (size_t)(row0 + j + half * 8) * HID;
        h0[r + 0  + m] = acc0[j];
        h0[r + 16 + m] = acc1[j];
        h0[r + 32 + m] = acc2[j];
        h0[r + 48 + m] = acc3[j];
    }
}

// ---------------- sparse aggregation ----------------
// h1[col[e], f] += h0[row[e], f] * norm[e]   (edge,feature) flattened: coalesced per row
__global__ void scatter_feat_kernel(const int* __restrict__ row, const int* __restrict__ col,
                                    const float* __restrict__ norm,
                                    const float* __restrict__ h0, float* __restrict__ h1,
                                    int total) {
    int t = blockIdx.x * blockDim.x + threadIdx.x;
    if (t >= total) return;
    int e = t >> 6;
    int f = t & 63;
    float v = h0[(size_t)row[e] * HID + f] * norm[e];
    atomicAdd(&h1[(size_t)col[e] * HID + f], v);
}

// h0[n,f] = relu(h1[n,f] + dinv[n]^2 * h0[n,f] + b1[f])   (self-loop folded in)
__global__ void combine_relu_kernel(const float* __restrict__ h1, const float* __restrict__ dinv,
                                    const float* __restrict__ b1, float* __restrict__ h0,
                                    int total) {
    int i = blockIdx.x * blockDim.x + threadIdx.x;
    if (i >= total) return;
    int n = i >> 6;
    int f = i & 63;
    float di = dinv[n];
    float v  = h1[i] + di * di * h0[i] + b1[f];
    h0[i] = fmaxf(v, 0.0f);
}

// z[n] = dot(h0[n,:], W2[:,0])
__global__ void dot_w2_kernel(const float* __restrict__ h, const float* __restrict__ W2,
                              float* __restrict__ z, int n) {
    int i = blockIdx.x * blockDim.x + threadIdx.x;
    if (i >= n) return;
    const float* __restrict__ hr = h + (size_t)i * HID;
    float s = 0.0f;
#pragma unroll
    for (int f = 0; f < HID; ++f) s = fmaf(hr[f], W2[f], s);
    z[i] = s;
}

// out[col[e]] += z[row[e]] * norm[e]
__global__ void scatter_scalar_kernel(const int* __restrict__ row, const int* __restrict__ col,
                                      const float* __restrict__ norm, const float* __restrict__ z,
                                      float* __restrict__ out, int e) {
    int i = blockIdx.x * blockDim.x + threadIdx.x;
    if (i < e) atomicAdd(&out[col[i]], z[row[i]] * norm[i]);
}

// out[n] += dinv[n]^2 * z[n] + b2
__global__ void finalize_kernel(float* __restrict__ out, const float* __restrict__ dinv,
                                const float* __restrict__ z, const float* __restrict__ b2,
                                int n) {
    int i = blockIdx.x * blockDim.x + threadIdx.x;
    if (i >= n) return;
    float di = dinv[i];
    out[i] += di * di * z[i] + b2[0];
}

// ---------------- launcher ----------------
extern "C" void kernel_launch(void* const* d_in, const int* in_sizes, int n_in,
                              void* d_out, int out_size, void* d_ws, size_t ws_size,
                              hipStream_t stream) {
    const float* x   = (const float*)d_in[0];
    const int*   ei  = (const int*)  d_in[1];
    const float* w   = (const float*)d_in[2];
    const float* W1  = (const float*)d_in[3];
    const float* b1  = (const float*)d_in[4];
    const float* W2  = (const float*)d_in[5];
    const float* b2  = (const float*)d_in[6];
    float*       out = (float*)d_out;

    const int* row = ei;            // edge_index[0,:]
    const int* col = ei + N_EDGES;  // edge_index[1,:]

    float* ws   = (float*)d_ws;
    float* dinv = ws;                                   // N
    float* norm = dinv + N_NODES;                       // E
    float* h0   = norm + N_EDGES;                       // N*64
    float* h1   = h0 + (size_t)N_NODES * HID;           // N*64
    float* z    = h1 + (size_t)N_NODES * HID;           // N

    const int B = 256;
    auto g = [](int n, int b) { return (n + b - 1) / b; };

    // normalization coefficients
    fill_kernel<<<g(N_NODES, B), B, 0, stream>>>(dinv, 1.0f, N_NODES);   // self-loop weight
    deg_kernel<<<g(N_EDGES, B), B, 0, stream>>>(col, w, dinv, N_EDGES);
    dinv_kernel<<<g(N_NODES, B), B, 0, stream>>>(dinv, N_NODES);
    norm_kernel<<<g(N_EDGES, B), B, 0, stream>>>(row, col, w, dinv, norm, N_EDGES);

    // zero accumulators
    fill_kernel<<<g(N_NODES * HID, B), B, 0, stream>>>(h1, 0.0f, N_NODES * HID);
    fill_kernel<<<g(N_NODES, B), B, 0, stream>>>(out, 0.0f, N_NODES);

    // conv1: dense (WMMA) then sparse aggregate + bias + relu
    {
        const int waves_per_block = 4;                  // 128 threads
        const int strips = N_NODES / 16;                // 6250
        gemm_wmma_kernel<<<g(strips, waves_per_block), waves_per_block * 32, 0, stream>>>(
            x, W1, h0, N_NODES);
    }
    scatter_feat_kernel<<<g(N_EDGES * HID, B), B, 0, stream>>>(row, col, norm, h0, h1,
                                                               N_EDGES * HID);
    combine_relu_kernel<<<g(N_NODES * HID, B), B, 0, stream>>>(h1, dinv, b1, h0,
                                                               N_NODES * HID);

    // conv2: [N,64]@[64,1] then scalar sparse aggregate
    dot_w2_kernel<<<g(N_NODES, B), B, 0, stream>>>(h0, W2, z, N_NODES);
    scatter_scalar_kernel<<<g(N_EDGES, B), B, 0, stream>>>(row, col, norm, z, out, N_EDGES);
    finalize_kernel<<<g(N_NODES, B), B, 0, stream>>>(out, dinv, z, b2, N_NODES);
}